// GAT_27410481283417
// MI455X (gfx1250) — compile-verified
//
#include <hip/hip_runtime.h>
#include <hip/hip_bf16.h>

#define N_V 50000
#define F_DIM 128
#define K_DIM 128
#define H_HEADS 3
#define M_NB 10

typedef __attribute__((ext_vector_type(2))) float v2f;
typedef __attribute__((ext_vector_type(8))) float v8f;

// ---------------------------------------------------------------------------
// Kernel A: out[h][n][k] = sum_f vertices[n][f] * W[h][f][k]
// One workgroup = one (head, matrix) combo x 128-row chunk.
// Weights staged in LDS (64 KB) in K-PAIR-INTERLEAVED layout:
//   sW[(p*128 + c)*2 + par] = W[2p + par][c]
// so each WMMA B-fragment (rows ka, ka+1 at one column) is a single aligned
// ds_load_b64 into an adjacent VGPR pair - no repack moves.
// Each wave produces a 16-row x 128-col strip via V_WMMA_F32_16X16X4_F32.
// ---------------------------------------------------------------------------
__global__ __launch_bounds__(256) void gat_gemm_wmma(
    const float* __restrict__ vertices,
    const float* __restrict__ Wvc,
    const float* __restrict__ Wint,
    const float* __restrict__ Wnh,
    float* __restrict__ Zc,
    float* __restrict__ Vint,
    float* __restrict__ Vnh)
{
    __shared__ float sW[F_DIM * K_DIM];   // 64 KB

    const int combo = blockIdx.y;         // 0..8
    const int h = combo / 3;
    const int w = combo % 3;
    const float* Wsrc = (w == 0 ? Wvc : (w == 1 ? Wint : Wnh))
                        + (size_t)h * F_DIM * K_DIM;
    float* Dst = (w == 0 ? Zc : (w == 1 ? Vint : Vnh))
                 + (size_t)h * N_V * K_DIM;

    const int tid = threadIdx.x;

    // Stage W[h] into LDS, interleaving row-pairs on the fly.
    // i indexes (pair, c): pair = i>>7, c = i&127 (4 consecutive columns).
    for (int i = tid * 4; i < (F_DIM / 2) * K_DIM; i += 256 * 4) {
        const int pair = i >> 7;
        const int c = i & 127;
        const float4 r0 = *(const float4*)(Wsrc + (size_t)(pair * 2) * K_DIM + c);
        const float4 r1 = *(const float4*)(Wsrc + (size_t)(pair * 2 + 1) * K_DIM + c);
        float2* base = (float2*)&sW[(pair * K_DIM + c) * 2];
        base[0] = make_float2(r0.x, r1.x);
        base[1] = make_float2(r0.y, r1.y);
        base[2] = make_float2(r0.z, r1.z);
        base[3] = make_float2(r0.w, r1.w);
    }
    __syncthreads();

    const int wave = tid >> 5;
    const int lane = tid & 31;
    const int lg = lane >> 4;             // lane group: 0 or 1
    const int lm = lane & 15;
    const int n0 = blockIdx.x * 128 + wave * 16;
    if (n0 >= N_V) return;                // wave-uniform (N_V % 16 == 0)

    const float* arow = vertices + (size_t)(n0 + lm) * F_DIM;

    v8f acc[8] = {};

    for (int kk = 0; kk < F_DIM; kk += 4) {
        const int ka = kk + lg * 2;       // K index of VGPR0 of the fragments
        // A fragment: lane holds M=lm, K=ka..ka+1 (8B, L0-resident row)
        const float2 av = *(const float2*)(arow + ka);
        v2f afrag;
        afrag.x = av.x;
        afrag.y = av.y;
        // B fragments: base + immediate offsets j*128B, one b64 each
        const float* bbase = &sW[(((kk >> 1) + lg) * K_DIM + lm) * 2];
#pragma unroll
        for (int j = 0; j < 8; ++j) {
            const float2 b2 = *(const float2*)(bbase + j * 32);
            v2f bfrag;
            bfrag.x = b2.x;
            bfrag.y = b2.y;
            acc[j] = __builtin_amdgcn_wmma_f32_16x16x4_f32(
                false, afrag, false, bfrag, (short)0, acc[j], false, false);
        }
    }

    // C layout: VGPR r -> M = r + lg*8, N = lm
#pragma unroll
    for (int j = 0; j < 8; ++j) {
#pragma unroll
        for (int r = 0; r < 8; ++r) {
            const int row = n0 + r + lg * 8;
            Dst[(size_t)row * K_DIM + j * 16 + lm] = acc[j][r];
        }
    }
}

// ---------------------------------------------------------------------------
// Kernel B: gather attention + combine. One wave32 per (h, n); each lane owns
// 4 k's. Gathers are coalesced 512 B row reads.
// ---------------------------------------------------------------------------
__device__ __forceinline__ float wave_sum32(float v) {
#pragma unroll
    for (int off = 16; off > 0; off >>= 1) v += __shfl_xor(v, off, 32);
    return v;
}

__device__ __forceinline__ float4 attn_branch(
    const float* __restrict__ Vh,        // V + h*N*K
    const int* __restrict__ idx_row,     // idx + n*M
    const float* __restrict__ edge_row,  // edges + n*M
    const float4 an4, const float ec, const int k0)
{
    float  e[M_NB];
    float4 g[M_NB];
    float  cnt = 0.0f;
#pragma unroll
    for (int m = 0; m < M_NB; ++m) {
        const int j = idx_row[m];        // wave-uniform
        if (j != -1) {
            g[m] = *(const float4*)(Vh + (size_t)j * K_DIM + k0);
            float d = g[m].x * an4.x + g[m].y * an4.y +
                      g[m].z * an4.z + g[m].w * an4.w;
            d = wave_sum32(d);
            e[m] = (d + ec) * edge_row[m];   // e * mask * edges (mask==1)
            cnt += 1.0f;
        } else {
            g[m] = make_float4(0.f, 0.f, 0.f, 0.f);  // g * mask
            e[m] = 0.0f;                              // masked e == 0, still in softmax
        }
    }
    // softmax over all M entries (masked ones included, as in the reference)
    float mx = e[0];
#pragma unroll
    for (int m = 1; m < M_NB; ++m) mx = fmaxf(mx, e[m]);
    float s = 0.0f;
    float wgt[M_NB];
#pragma unroll
    for (int m = 0; m < M_NB; ++m) { wgt[m] = __expf(e[m] - mx); s += wgt[m]; }
    const float inv = 1.0f / (s * fmaxf(cnt, 1.0f));  // softmax denom * norm
    float4 zn = make_float4(0.f, 0.f, 0.f, 0.f);
#pragma unroll
    for (int m = 0; m < M_NB; ++m) {
        const float al = wgt[m] * inv;
        zn.x += g[m].x * al; zn.y += g[m].y * al;
        zn.z += g[m].z * al; zn.w += g[m].w * al;
    }
    return zn;
}

__global__ __launch_bounds__(256) void gat_attn(
    const float* __restrict__ Zc, const float* __restrict__ Vint,
    const float* __restrict__ Vnh,
    const int* __restrict__ int_idx, const int* __restrict__ nh_idx,
    const float* __restrict__ int_edges, const float* __restrict__ nh_edges,
    const float* __restrict__ avec, const float* __restrict__ bv,
    float* __restrict__ out)
{
    const int gw = (blockIdx.x * 256 + threadIdx.x) >> 5;  // wave id
    if (gw >= H_HEADS * N_V) return;
    const int h = gw / N_V;
    const int n = gw - h * N_V;
    const int lane = threadIdx.x & 31;
    const int k0 = lane * 4;

    const float* an = avec + (size_t)h * 2 * K_DIM;        // a[h,:,0]
    const float4 an4 = *(const float4*)(an + k0);
    const float4 ac4 = *(const float4*)(an + K_DIM + k0);
    const float4 zc  = *(const float4*)(Zc + ((size_t)h * N_V + n) * K_DIM + k0);

    float ec = zc.x * ac4.x + zc.y * ac4.y + zc.z * ac4.z + zc.w * ac4.w;
    ec = wave_sum32(ec);                                   // e_center[h][n]

    const float4 zi = attn_branch(Vint + (size_t)h * N_V * K_DIM,
                                  int_idx + (size_t)n * M_NB,
                                  int_edges + (size_t)n * M_NB, an4, ec, k0);
    const float4 zh = attn_branch(Vnh + (size_t)h * N_V * K_DIM,
                                  nh_idx + (size_t)n * M_NB,
                                  nh_edges + (size_t)n * M_NB, an4, ec, k0);
    const float4 b4 = *(const float4*)(bv + (size_t)h * K_DIM + k0);

    float4 z;
    z.x = fmaxf(zc.x + zi.x + zh.x + b4.x, 0.0f);
    z.y = fmaxf(zc.y + zi.y + zh.y + b4.y, 0.0f);
    z.z = fmaxf(zc.z + zi.z + zh.z + b4.z, 0.0f);
    z.w = fmaxf(zc.w + zi.w + zh.w + b4.w, 0.0f);

    *(float4*)(out + (size_t)n * (H_HEADS * K_DIM) + h * K_DIM + k0) = z;
}

// ---------------------------------------------------------------------------
extern "C" void kernel_launch(void* const* d_in, const int* in_sizes, int n_in,
                              void* d_out, int out_size, void* d_ws, size_t ws_size,
                              hipStream_t stream)
{
    const float* vertices    = (const float*)d_in[0];
    const int*   nh_indices  = (const int*)  d_in[1];
    const int*   int_indices = (const int*)  d_in[2];
    const float* nh_edges    = (const float*)d_in[3];
    const float* int_edges   = (const float*)d_in[4];
    // d_in[5] = is_int : unused by the reference
    const float* Wvc         = (const float*)d_in[6];
    const float* bv          = (const float*)d_in[7];
    const float* Wvn_int     = (const float*)d_in[8];
    const float* Wvn_nh      = (const float*)d_in[9];
    const float* a           = (const float*)d_in[10];

    // Workspace: Zc | Vint | Vnh, each H*N*K fp32 (76.8 MB, 230.4 MB total)
    float* Zc   = (float*)d_ws;
    float* Vint = Zc   + (size_t)H_HEADS * N_V * K_DIM;
    float* Vnh  = Vint + (size_t)H_HEADS * N_V * K_DIM;

    dim3 gridA((N_V + 127) / 128, H_HEADS * 3);
    gat_gemm_wmma<<<gridA, 256, 0, stream>>>(vertices, Wvc, Wvn_int, Wvn_nh,
                                             Zc, Vint, Vnh);

    const int nwaves = H_HEADS * N_V;                 // 150000
    gat_attn<<<(nwaves + 7) / 8, 256, 0, stream>>>(Zc, Vint, Vnh,
                                                   int_indices, nh_indices,
                                                   int_edges, nh_edges,
                                                   a, bv, (float*)d_out);
}